// PathDevelopmentNetwork_83537113907800
// MI455X (gfx1250) — compile-verified
//
#include <hip/hip_runtime.h>

#define T_LEN    8192
#define CHUNK    2048
#define NCHUNK   (T_LEN / CHUNK)
#define THREADS  256
#define KT       4
#define D_CH     8
#define A_CONST  (-50.0f)

typedef float v2f __attribute__((ext_vector_type(2)));
typedef float v8f __attribute__((ext_vector_type(8)));

// Per-wave 256-element (16x16) inclusive scan via V_WMMA_F32_16X16X4_F32:
//   P = X * U  (U = upper-triangular ones, K=16 split into 4 chained K=4 WMMAs)
//   row carries from P(:,15) via tiny exclusive scan in LDS,
//   cross-wave / cross-chunk carry via LDS totals.
template <typename F>
__device__ __forceinline__ void tile_scan_wmma(
    F value, float* __restrict__ dst,
    float (*rs)[16], float* tot, float* carry, int qidx, int tid)
{
  const int wave  = tid >> 5;
  const int lane  = tid & 31;
  const int hi    = lane >> 4;   // which half-wave
  const int col   = lane & 15;   // A: row index (M). B/C/D: column index (N)
  const int wbase = wave * 256;

  v8f acc = {0.f, 0.f, 0.f, 0.f, 0.f, 0.f, 0.f, 0.f};
#pragma unroll
  for (int q = 0; q < 4; ++q) {
    const int k0 = 4 * q + hi * 2;          // K columns held by this half-wave
    v2f a, b;
    a.x = value(wbase + 16 * col + k0);     // X(M=col, K=k0)
    a.y = value(wbase + 16 * col + k0 + 1); // X(M=col, K=k0+1)
    b.x = (k0     <= col) ? 1.f : 0.f;      // U(K=k0,   N=col)
    b.y = (k0 + 1 <= col) ? 1.f : 0.f;      // U(K=k0+1, N=col)
    acc = __builtin_amdgcn_wmma_f32_16x16x4_f32(false, a, false, b,
                                                (short)0, acc, false, false);
  }
  // Last column of P = per-row sums of the tile.
  if (col == 15) {
#pragma unroll
    for (int v = 0; v < 8; ++v) rs[wave][hi * 8 + v] = acc[v];
  }
  __syncthreads();
  // Exclusive prefix over this wave's 16 row sums; run == tile total.
  float e[8];
  float run = 0.f;
#pragma unroll
  for (int r = 0; r < 16; ++r) {
    if ((r >> 3) == hi) e[r & 7] = run;
    run += rs[wave][r];
  }
  if (lane == 0) tot[wave] = run;
  __syncthreads();
  float cbase = carry[qidx];
  for (int wv = 0; wv < wave; ++wv) cbase += tot[wv];
  // Write globally-corrected inclusive scan back to LDS (linear layout).
#pragma unroll
  for (int v = 0; v < 8; ++v) {
    const int r = hi * 8 + v;
    dst[wbase + 16 * r + col] = acc[v] + e[v] + cbase;
  }
  __syncthreads();
  if (tid == 0) {
    float s = 0.f;
    for (int wv = 0; wv < 8; ++wv) s += tot[wv];
    carry[qidx] += s;
  }
  __syncthreads();
}

__global__ __launch_bounds__(THREADS)
void pathdev_fused_kernel(const float* __restrict__ x,
                          const float* __restrict__ w1,
                          const float* __restrict__ b1,
                          const float* __restrict__ w2,
                          const float* __restrict__ b2,
                          float* __restrict__ out,
                          int words)
{
  __shared__ float s_dx1[CHUNK];
  __shared__ float s_dx2[CHUNK];
  __shared__ float s_c2[CHUNK];      // inclusive cumsum(dx2)
  __shared__ float s_d1[CHUNK];      // inclusive cumsum(dx1*dx2)
  __shared__ float s_rs[8][16];
  __shared__ float s_tot[8];
  __shared__ float s_carry[2];
  __shared__ float s_headc1[KT];     // c1[0..KT-1]
  __shared__ float s_headiss[KT];    // cumsum(dx2*c1_prev)[0..KT-1]
  __shared__ float s_taps[4][KT];    // f, fp, g, gp

  const int tid = threadIdx.x;
  const int row = blockIdx.x;        // (b, w) pair
  const int b   = row / words;
  const int w   = row % words;

  const float* dx1 = x + ((size_t)b * D_CH + (w % D_CH)) * T_LEN;
  const float* dx2 = x + ((size_t)b * D_CH + ((w + 1) % D_CH)) * T_LEN;
  float* orow = out + (size_t)row * (T_LEN + 1);

  if (tid == 0) {
    s_carry[0] = 0.f;
    s_carry[1] = 0.f;
    // FNN taps: taps[k][t] = (sum_h w2[k,0,h]*tanh(t*w1[k,h,0]+b1[k,h]) + b2[k]) * exp(-50 t)
    // exp(-50t) underflows to 0.0f for t >= 3, so KT=4 taps are fp32-exact.
    for (int k = 0; k < 4; ++k) {
      for (int t = 0; t < KT; ++t) {
        float s = b2[k];
        for (int h = 0; h < 5; ++h)
          s += w2[k * 5 + h] * tanhf((float)t * w1[k * 5 + h] + b1[k * 5 + h]);
        s_taps[k][t] = s * expf(A_CONST * (float)t);
      }
    }
  }

  for (int c = 0; c < NCHUNK; ++c) {
    const int base = c * CHUNK;
    const float* g1 = dx1 + base;
    const float* g2 = dx2 + base;

    // Stage the chunk with the CDNA5 async LDS engine (ASYNCcnt path):
    // each lane issues B128 loads straight into LDS, no VGPR round-trip.
#pragma unroll
    for (int half = 0; half < 2; ++half) {
      const int fo = half * (CHUNK / 2) + tid * 4;              // float offset
      const unsigned l1 = (unsigned)(size_t)&s_dx1[fo];         // LDS byte addr
      const unsigned l2 = (unsigned)(size_t)&s_dx2[fo];
      const unsigned long long a1 = (unsigned long long)(g1 + fo);
      const unsigned long long a2 = (unsigned long long)(g2 + fo);
      asm volatile("global_load_async_to_lds_b128 %0, %1, off"
                   :: "v"(l1), "v"(a1) : "memory");
      asm volatile("global_load_async_to_lds_b128 %0, %1, off"
                   :: "v"(l2), "v"(a2) : "memory");
    }
    // Prefetch next chunk into L2 while this chunk is being scanned.
    if (c + 1 < NCHUNK) {
      __builtin_prefetch(&g1[CHUNK + tid * 8], 0, 1);
      __builtin_prefetch(&g2[CHUNK + tid * 8], 0, 1);
    }
    asm volatile("s_wait_asynccnt 0x0" ::: "memory");
    __syncthreads();

    if (c == 0 && tid == 0) {
      // Heads: only the first KT values of cs_d1 / iss_sum are ever multiplied
      // by a nonzero (reversed-kernel) factor.
      float c1 = 0.f, iss = 0.f;
      for (int j = 0; j < KT; ++j) {
        iss += s_dx2[j] * c1;        // c1 currently holds c1[j-1]
        s_headiss[j] = iss;
        c1 += s_dx1[j];
        s_headc1[j] = c1;
      }
    }

    // Full-sequence scans (the serial dependency) via WMMA.
    tile_scan_wmma([&](int e) { return s_dx2[e]; },
                   s_c2, s_rs, s_tot, s_carry, 0, tid);
    tile_scan_wmma([&](int e) { return s_dx1[e] * s_dx2[e]; },
                   s_d1, s_rs, s_tot, s_carry, 1, tid);

    // Outputs m in [base+1, base+CHUNK]. Every one of the six products has a
    // reversed-kernel factor => out[m] == 0 for m <= T-KT.
#pragma unroll
    for (int r = 0; r < CHUNK / THREADS; ++r) {
      const int gm = base + 1 + tid + r * THREADS;
      float val = 0.f;
      if (gm >= T_LEN - KT + 1) {
        const float* f  = s_taps[0];
        const float* fp = s_taps[1];
        const float* g  = s_taps[2];
        const float* gp = s_taps[3];
        float AB = 0.f, BB = 0.f, CB = 0.f, DB = 0.f, EB = 0.f, FB = 0.f;
#pragma unroll
        for (int a = 0; a < KT; ++a) {
          AB += g[a];
          BB += gp[a];
          const int idx = gm - 1 - a;        // inside this (last) chunk here
          const float c2v = s_c2[idx - base];
          const float d1v = s_d1[idx - base];
          CB += g[a] * c2v;  DB += gp[a] * c2v;
          EB += g[a] * d1v;  FB += gp[a] * d1v;
        }
        float AA = 0.f, BA = 0.f, CA = 0.f, DA = 0.f, EA = 0.f, FA = 0.f;
#pragma unroll
        for (int tau = 0; tau < KT; ++tau) {
          const int mp = gm - T_LEN + tau;   // index into pre0(head) arrays
          if (mp >= 0) {
            const float issv = (mp == 0) ? 0.f : s_headiss[mp - 1];
            const float c1v  = (mp == 0) ? 0.f : s_headc1[mp - 1];
            AA += f[tau] * issv;  BA += fp[tau] * issv;
            CA += f[tau] * c1v;   DA += fp[tau] * c1v;
            EA += f[tau];         FA += fp[tau];
          }
        }
        val = 0.5f * (AA * AB + BA * BB + CA * CB + DA * DB + EA * EB + FA * FB);
      }
      orow[gm] = val;
    }
    if (c == 0 && tid == 0) orow[0] = 0.f;
    __syncthreads();
  }
}

extern "C" void kernel_launch(void* const* d_in, const int* in_sizes, int n_in,
                              void* d_out, int out_size, void* d_ws, size_t ws_size,
                              hipStream_t stream) {
  (void)n_in; (void)d_ws; (void)ws_size;
  const float* x  = (const float*)d_in[0];
  const float* w1 = (const float*)d_in[1];
  const float* b1 = (const float*)d_in[2];
  const float* w2 = (const float*)d_in[3];
  const float* b2 = (const float*)d_in[4];
  const int bs    = in_sizes[0] / (D_CH * T_LEN);       // 32
  const int words = out_size / (bs * (T_LEN + 1));      // 8
  dim3 grid(bs * words), block(THREADS);
  hipLaunchKernelGGL(pathdev_fused_kernel, grid, block, 0, stream,
                     x, w1, b1, w2, b2, (float*)d_out, words);
}